// CounterexampleGenerator_53446573031674
// MI455X (gfx1250) — compile-verified
//
#include <hip/hip_runtime.h>
#include <hip/hip_bf16.h>
#include <stdint.h>

// Problem constants (match reference)
#define Bc 64
#define Lc 1024
#define Dc 2048
#define Hc 128
#define NSPL 8          // L-splits for pooling parallelism (64*2*8 = 1024 blocks)
#define PGD_STEP 0.01f
#define PGD_EPS  0.2f

typedef __attribute__((ext_vector_type(2))) float v2f;
typedef __attribute__((ext_vector_type(4))) float v4f;
typedef __attribute__((ext_vector_type(8))) float v8f;

// ---------------------------------------------------------------------------
// Kernel 1: fused stream — read x once, write copy to d_out, emit partial sums
// over an L-slice. 1024 blocks x 256 threads, B128 non-temporal traffic (x is
// touched exactly once; keep the 192MB L2 clean for the PGD working set).
// ---------------------------------------------------------------------------
__global__ __launch_bounds__(256)
void ceg_pool_copy(const float* __restrict__ x, float* __restrict__ xout,
                   float* __restrict__ partial) {
  const int tid = threadIdx.x;
  const int bid = blockIdx.x;
  const int ls  = bid & (NSPL - 1);         // L-split index
  const int dch = (bid >> 3) & 1;           // D chunk (2 x 1024 floats)
  const int b   = bid >> 4;                 // batch
  const int d0  = dch * 1024 + tid * 4;
  const int l0  = ls * (Lc / NSPL);

  v4f acc = {0.f, 0.f, 0.f, 0.f};
#pragma unroll 8
  for (int l = l0; l < l0 + (Lc / NSPL); ++l) {
    const size_t idx = ((size_t)b * Lc + l) * Dc + d0;
    v4f v = __builtin_nontemporal_load((const v4f*)(x + idx));   // TH_NT load
    acc += v;
    __builtin_nontemporal_store(v, (v4f*)(xout + idx));          // TH_NT store
  }
  const size_t pidx = ((size_t)ls * Bc + b) * Dc + d0;
  *(v4f*)(partial + pidx) = acc;            // small, reused next kernel: RT
}

// ---------------------------------------------------------------------------
// Kernel 2: reduce partials -> pooled, add deterministic pseudo-normal noise
// -> x_adv0.  (JAX threefry not reproducible; PCG hash + Box-Muller instead.)
// ---------------------------------------------------------------------------
__device__ __forceinline__ uint32_t ceg_pcg(uint32_t v) {
  v = v * 747796405u + 2891336453u;
  uint32_t w = ((v >> ((v >> 28u) + 4u)) ^ v) * 277803737u;
  return (w >> 22u) ^ w;
}

__global__ __launch_bounds__(256)
void ceg_init(const float* __restrict__ partial, float* __restrict__ pooled,
              float* __restrict__ xadv) {
  const int i = blockIdx.x * 256 + threadIdx.x;     // < B*D
  float s = 0.f;
#pragma unroll
  for (int ls = 0; ls < NSPL; ++ls)
    s += __builtin_nontemporal_load(partial + (size_t)ls * Bc * Dc + i);
  const float p = s * (1.0f / (float)Lc);

  const uint32_t h1 = ceg_pcg(0x9E3779B9u ^ (uint32_t)i);
  const uint32_t h2 = ceg_pcg(h1 + 0x85EBCA6Bu);
  const float u1 = ((float)h1 + 0.5f) * (1.0f / 4294967296.0f);
  const float u2 = ((float)h2 + 0.5f) * (1.0f / 4294967296.0f);
  const float n  = sqrtf(-2.0f * logf(u1)) * cosf(6.28318530718f * u2);

  pooled[i] = p;
  xadv[i]   = p + 0.01f * n;
}

// ---------------------------------------------------------------------------
// Kernel 3: forward pass via V_WMMA_F32_16X16X4_F32.
//   z1 = x_adv @ W1 + b1  -> gelu -> dot with W2 -> sigmoid
// Grid: 4 blocks (one 16-row M tile each), 8 waves (one 16-col N tile each,
// covering the full H=128 so the per-row reduction stays in LDS).
// mode==0: emit wvec[b,h] = s(1-s) * gelu'(z1) * W2[h]   (for backward)
// mode==1: emit worst_score / cert_score to d_out
// ---------------------------------------------------------------------------
__global__ __launch_bounds__(256)
void ceg_fwd(const float* __restrict__ xadv, const float* __restrict__ W1,
             const float* __restrict__ b1, const float* __restrict__ W2,
             const float* __restrict__ b2, float* __restrict__ wvec,
             float* __restrict__ worst, float* __restrict__ cert, int mode) {
  __shared__ float tmp[16][Hc];   // gelu'(z1)*W2 staging
  __shared__ float z2sum[16];
  __shared__ float coefS[16];

  const int tid  = threadIdx.x;
  const int wave = tid >> 5;
  const int lane = tid & 31;
  const int lh   = lane & 15;
  const int hi   = lane >> 4;
  const int m0   = blockIdx.x * 16;
  const int n0   = wave * 16;

  if (tid < 16) z2sum[tid] = 0.f;
  __syncthreads();

  v8f acc = {0.f, 0.f, 0.f, 0.f, 0.f, 0.f, 0.f, 0.f};
  const float* arow = xadv + (size_t)(m0 + lh) * Dc;
#pragma unroll 4
  for (int k0 = 0; k0 < Dc; k0 += 4) {
    const int kb = k0 + 2 * hi;
    v2f a = *(const v2f*)(arow + kb);                 // A: x_adv 16x4 slice
    v2f bv;
    bv.x = W1[(size_t)kb * Hc + n0 + lh];             // B: W1 4x16 slice
    bv.y = W1[(size_t)(kb + 1) * Hc + n0 + lh];
    acc = __builtin_amdgcn_wmma_f32_16x16x4_f32(
        false, a, false, bv, (short)0, acc, false, false);
  }

  const int   col = n0 + lh;
  const float w2c = W2[col];
  const float bb1 = b1[col];
#pragma unroll
  for (int i = 0; i < 8; ++i) {
    const int   r = i + 8 * hi;                       // tile-local row
    const float z = acc[i] + bb1;
    const float e = erff(z * 0.70710678118654752f);   // exact GELU
    const float g = 0.5f * z * (1.f + e);
    const float dg = 0.5f * (1.f + e) +
                     z * 0.39894228040143268f * expf(-0.5f * z * z);
    atomicAdd(&z2sum[r], g * w2c);                    // ds_add_f32
    tmp[r][col] = dg * w2c;
  }
  __syncthreads();

  if (tid < 16) {
    const float z = z2sum[tid] + b2[0];
    const float s = 1.f / (1.f + expf(-z));
    coefS[tid] = s * (1.f - s);
    if (mode) { worst[m0 + tid] = s; cert[m0 + tid] = 1.f - s; }
  }
  __syncthreads();

  if (!mode) {
#pragma unroll
    for (int i = 0; i < 8; ++i) {
      const int r = i + 8 * hi;
      wvec[(size_t)(m0 + r) * Hc + col] = coefS[r] * tmp[r][col];
    }
  }
}

// ---------------------------------------------------------------------------
// Kernel 4: backward + PGD update via WMMA.
//   g[b,d] = sum_h wvec[b,h] * W1[d,h]   (B operand is contiguous f32x2 load)
//   x_adv  = pooled + clip(x_adv - step*sign(g) - pooled, +-eps)
// 512 tiles (4 M x 128 N over D) -> 64 blocks x 8 waves.
// ---------------------------------------------------------------------------
__global__ __launch_bounds__(256)
void ceg_bwd(const float* __restrict__ wvec, const float* __restrict__ W1,
             const float* __restrict__ pooled, float* __restrict__ xadv) {
  const int tid  = threadIdx.x;
  const int wave = tid >> 5;
  const int lane = tid & 31;
  const int lh   = lane & 15;
  const int hi   = lane >> 4;

  const int tile = blockIdx.x * 8 + wave;   // 0..511
  const int m0   = (tile >> 7) * 16;        // batch tile
  const int n0   = (tile & 127) * 16;       // D tile

  v8f acc = {0.f, 0.f, 0.f, 0.f, 0.f, 0.f, 0.f, 0.f};
  const float* arow = wvec + (size_t)(m0 + lh) * Hc;
  const float* brow = W1 + (size_t)(n0 + lh) * Hc;  // W1[d,:] row, k indexes h
#pragma unroll 4
  for (int k0 = 0; k0 < Hc; k0 += 4) {
    const int kb = k0 + 2 * hi;
    v2f a = *(const v2f*)(arow + kb);
    v2f b = *(const v2f*)(brow + kb);
    acc = __builtin_amdgcn_wmma_f32_16x16x4_f32(
        false, a, false, b, (short)0, acc, false, false);
  }

#pragma unroll
  for (int i = 0; i < 8; ++i) {
    const int    r   = i + 8 * hi;
    const size_t idx = (size_t)(m0 + r) * Dc + n0 + lh;
    const float  g   = acc[i];
    const float  sg  = (g > 0.f) ? 1.f : ((g < 0.f) ? -1.f : 0.f);
    const float  p   = pooled[idx];
    float dlt = (xadv[idx] - PGD_STEP * sg) - p;
    dlt = fminf(PGD_EPS, fmaxf(-PGD_EPS, dlt));
    xadv[idx] = p + dlt;
  }
}

// ---------------------------------------------------------------------------
// Kernel 5: violated = min(worst_score) < 0.1  (64 elements; trivial)
// ---------------------------------------------------------------------------
__global__ void ceg_violated(const float* __restrict__ worst,
                             float* __restrict__ outv) {
  if (threadIdx.x == 0) {
    float mn = worst[0];
    for (int i = 1; i < Bc; ++i) mn = fminf(mn, worst[i]);
    outv[0] = (mn < 0.1f) ? 1.0f : 0.0f;
  }
}

// ---------------------------------------------------------------------------
extern "C" void kernel_launch(void* const* d_in, const int* in_sizes, int n_in,
                              void* d_out, int out_size, void* d_ws,
                              size_t ws_size, hipStream_t stream) {
  const float* x  = (const float*)d_in[0];
  const float* W1 = (const float*)d_in[1];
  const float* b1 = (const float*)d_in[2];
  const float* W2 = (const float*)d_in[3];
  const float* b2 = (const float*)d_in[4];

  float* out = (float*)d_out;
  float* ws  = (float*)d_ws;

  // Workspace layout (floats)
  float* partial = ws;                                   // NSPL*B*D = 1,048,576
  float* pooled  = partial + (size_t)NSPL * Bc * Dc;     // B*D
  float* xadv    = pooled + (size_t)Bc * Dc;             // B*D
  float* wvec    = xadv + (size_t)Bc * Dc;               // B*H

  // Output layout: x copy | worst[64] | cert[64] | violated[1]
  float* xcopy = out;
  float* worst = out + (size_t)Bc * Lc * Dc;
  float* cert  = worst + Bc;
  float* viol  = cert + Bc;

  // 1) stream x: copy to output + L-split partial sums (bandwidth-bound phase)
  ceg_pool_copy<<<dim3(Bc * 2 * NSPL), dim3(256), 0, stream>>>(x, xcopy, partial);
  // 2) reduce -> pooled, add noise -> x_adv0
  ceg_init<<<dim3((Bc * Dc) / 256), dim3(256), 0, stream>>>(partial, pooled, xadv);
  // 3) PGD loop (10 iterations), WMMA fp32 GEMMs
  for (int it = 0; it < 10; ++it) {
    ceg_fwd<<<dim3(4), dim3(256), 0, stream>>>(xadv, W1, b1, W2, b2, wvec,
                                               nullptr, nullptr, 0);
    ceg_bwd<<<dim3(64), dim3(256), 0, stream>>>(wvec, W1, pooled, xadv);
  }
  // 4) final scoring -> worst/cert, then violated flag
  ceg_fwd<<<dim3(4), dim3(256), 0, stream>>>(xadv, W1, b1, W2, b2, wvec,
                                             worst, cert, 1);
  ceg_violated<<<dim3(1), dim3(32), 0, stream>>>(worst, viol);
}